// ColBERT_1726576854716
// MI455X (gfx1250) — compile-verified
//
#include <hip/hip_runtime.h>

typedef __attribute__((ext_vector_type(2))) float v2f;
typedef __attribute__((ext_vector_type(8))) float v8f;

#define QLEN   32
#define DIM    128
#define DLEN   220
#define NWAY   8
#define MTILES 14   // ceil(220/16)

__global__ __launch_bounds__(64) void colbert_maxsim_kernel(
    const float* __restrict__ Q, const float* __restrict__ D,
    const int* __restrict__ Dmask, float* __restrict__ out)
{
    const int doc  = blockIdx.x;            // 0 .. B*NWAY-1
    const int tid  = threadIdx.x;
    const int wave = tid >> 5;              // 2 waves / block
    const int lane = tid & 31;
    const int hi   = lane >> 4;             // half-wave select
    const int lo   = lane & 15;

    const float* qbase = Q + (size_t)(doc / NWAY) * QLEN * DIM;
    const float* dbase = D + (size_t)doc * DLEN * DIM;
    const int*   mbase = Dmask + (size_t)doc * DLEN;

    const int n0 = wave * 16;               // query-token tile base

    // ---- Preload B fragments (Q tile, transposed use) for all 32 K-steps ----
    // V_WMMA_F32_16X16X4_F32 B layout (4x16 KxN):
    //   lane(lo,hi): VGPR0 = B[K=2*hi+0, N=lo], VGPR1 = B[K=2*hi+1, N=lo]
    //   B[k][n] = Q[q=n0+n][k]  -> per lane one b64 load per K-step.
    v2f bfrag[32];
    const float* qrow = qbase + (size_t)(n0 + lo) * DIM + 2 * hi;
#pragma unroll
    for (int s = 0; s < 32; ++s)
        bfrag[s] = *(const v2f*)(qrow + 4 * s);

    float runmax = -9999.0f;

    for (int t = 0; t < MTILES; ++t) {
        const int m0 = t * 16;
        int arow = m0 + lo;
        if (arow > DLEN - 1) arow = DLEN - 1;   // clamp OOB rows; masked below
        const float* drow = dbase + (size_t)arow * DIM + 2 * hi;

        // prefetch next tile's row while this tile computes
        if (t + 1 < MTILES) {
            int prow = m0 + 16 + lo;
            if (prow > DLEN - 1) prow = DLEN - 1;
            __builtin_prefetch(dbase + (size_t)prow * DIM + 2 * hi, 0, 3);
        }

        v8f c = {0.f, 0.f, 0.f, 0.f, 0.f, 0.f, 0.f, 0.f};
#pragma unroll
        for (int s = 0; s < 32; ++s) {
            // A layout (16x4 MxK): lane(lo,hi): VGPR0 = A[M=lo, K=4s+2*hi],
            //                                   VGPR1 = A[M=lo, K=4s+2*hi+1]
            v2f a = *(const v2f*)(drow + 4 * s);
            c = __builtin_amdgcn_wmma_f32_16x16x4_f32(
                    /*neg_a=*/false, a, /*neg_b=*/false, bfrag[s],
                    /*c_mod=*/(short)0, c, /*reuse_a=*/false, /*reuse_b=*/false);
        }

        // C layout: VGPR v, lane(lo,hi) -> score[doctok = m0+v+8*hi][q = n0+lo]
#pragma unroll
        for (int v = 0; v < 8; ++v) {
            int   row = m0 + v + 8 * hi;
            int   mk  = (row < DLEN) ? mbase[row] : 0;
            float sc  = (mk > 0) ? c[v] : -9999.0f;
            runmax = fmaxf(runmax, sc);
        }
    }

    // lanes L and L^16 cover complementary row groups of the same q = n0+lo
    runmax = fmaxf(runmax, __shfl_xor(runmax, 16, 32));

    // sum the 16 per-q maxima of this wave's q-tile
    float ssum = runmax;
    ssum += __shfl_xor(ssum, 8, 32);
    ssum += __shfl_xor(ssum, 4, 32);
    ssum += __shfl_xor(ssum, 2, 32);
    ssum += __shfl_xor(ssum, 1, 32);

    __shared__ float partial[2];
    if (lane == 0) partial[wave] = ssum;
    __syncthreads();
    if (tid == 0) out[doc] = partial[0] + partial[1];
}

extern "C" void kernel_launch(void* const* d_in, const int* in_sizes, int n_in,
                              void* d_out, int out_size, void* d_ws, size_t ws_size,
                              hipStream_t stream) {
    const float* Q     = (const float*)d_in[0];
    const float* D     = (const float*)d_in[1];
    const int*   Dmask = (const int*)d_in[2];
    float*       out   = (float*)d_out;

    const int numDocs = out_size;           // B * NWAY = 1024
    colbert_maxsim_kernel<<<numDocs, 64, 0, stream>>>(Q, D, Dmask, out);
}